// psm_query_54185307406444
// MI455X (gfx1250) — compile-verified
//
#include <hip/hip_runtime.h>
#include <stdint.h>

#ifndef __has_builtin
#define __has_builtin(x) 0
#endif

#define BDIM 4
#define LDIM 5
#define CDIM 256
#define HWN  25200   // 100*252
#define HWN4 6300    // HWN/4
#define KSEL 1024

typedef float    f32x4 __attribute__((ext_vector_type(4)));
typedef unsigned u32x4 __attribute__((ext_vector_type(4)));

// Order-preserving float<->uint mapping (monotone, injective).
__device__ __forceinline__ unsigned enc_f(float f) {
  unsigned u = __float_as_uint(f);
  return (u & 0x80000000u) ? ~u : (u | 0x80000000u);
}
__device__ __forceinline__ float dec_f(unsigned u) {
  return __uint_as_float((u & 0x80000000u) ? (u & 0x7fffffffu) : ~u);
}

// ---------------------------------------------------------------------------
// Kernel 1: one block per (b,l). Computes d[hw] = max_cp (psm[b,l,cp,hw] -
// psm[b,0,cp,hw]) into 100.8KB of LDS (CDNA5: 320KB/WGP makes this resident),
// radix-selects the exact bit pattern of the K-th largest value, then writes
// the byte keep-mask fm[bl*HWN + hw] in {0,1}.
// ---------------------------------------------------------------------------
__global__ __launch_bounds__(256) void psm_thresh_kernel(
    const float* __restrict__ psm, unsigned char* __restrict__ fm) {
  __shared__ __align__(16) unsigned sh_d[HWN];   // 100800 B
  __shared__ unsigned sh_hist[256];
  __shared__ unsigned sh_prefix;
  __shared__ int sh_k;

  const int tid = threadIdx.x;
  const int bl  = blockIdx.x;            // b*L + l
  const int b   = bl / LDIM;

  const f32x4* __restrict__ pl = (const f32x4*)psm + (size_t)bl * 2 * HWN4;
  const f32x4* __restrict__ p0 = (const f32x4*)psm + (size_t)(b * LDIM) * 2 * HWN4;

  for (int i = tid; i < HWN4; i += 256) {
    f32x4 a0 = pl[i];            // cp = 0 row of (b,l)
    f32x4 e0 = p0[i];            // cp = 0 row of (b,0)
    f32x4 a1 = pl[i + HWN4];     // cp = 1
    f32x4 e1 = p0[i + HWN4];
    u32x4 u;
    u.x = enc_f(fmaxf(a0.x - e0.x, a1.x - e1.x));
    u.y = enc_f(fmaxf(a0.y - e0.y, a1.y - e1.y));
    u.z = enc_f(fmaxf(a0.z - e0.z, a1.z - e1.z));
    u.w = enc_f(fmaxf(a0.w - e0.w, a1.w - e1.w));
    *(u32x4*)(&sh_d[i * 4]) = u;   // ds_store_b128
  }
  if (tid == 0) sh_k = KSEL;
  __syncthreads();

  // MSB-first radix select: after 4 passes `prefix` is the exact 32-bit
  // pattern of the K-th largest mapped value.
  unsigned prefix = 0u, pmask = 0u;
  for (int pass = 0; pass < 4; ++pass) {
    const int shift = 24 - 8 * pass;
    sh_hist[tid] = 0u;           // blockDim == 256 == #bins
    __syncthreads();
    for (int i = tid; i < HWN; i += 256) {
      unsigned u = sh_d[i];
      if ((u & pmask) == prefix) atomicAdd(&sh_hist[(u >> shift) & 255u], 1u);
    }
    __syncthreads();
    if (tid == 0) {
      int k = sh_k, acc = 0, bin = 0;
      for (int bq = 255; bq >= 0; --bq) {
        int c = (int)sh_hist[bq];
        if (acc + c >= k) { bin = bq; break; }
        acc += c;
      }
      sh_k = k - acc;
      sh_prefix = prefix | ((unsigned)bin << shift);
    }
    __syncthreads();
    prefix = sh_prefix;
    pmask |= (255u << shift);
  }

  // Compare in *float* domain (matches reference semantics incl. +/-0 ties).
  const float th = dec_f(prefix);
  unsigned* __restrict__ fout = (unsigned*)(fm + (size_t)bl * HWN);
  for (int i = tid; i < HWN4; i += 256) {
    const u32x4 u = *(const u32x4*)(&sh_d[i * 4]);
    unsigned m = (dec_f(u.x) >= th ? 1u : 0u)
               | (dec_f(u.y) >= th ? 0x100u : 0u)
               | (dec_f(u.z) >= th ? 0x10000u : 0u)
               | (dec_f(u.w) >= th ? 0x1000000u : 0u);
    fout[i] = m;
  }
}

// ---------------------------------------------------------------------------
// Kernel 2: pure streaming apply. grid = (25, C=256, B*L=20), block = 256.
// Each thread handles one float4 of one (bl,c) row. Branches are uniform per
// block (bl is blockIdx.z). Output is never re-read -> non-temporal hints.
// The fm byte chunk is staged via the CDNA5 async global->LDS copy path.
// ---------------------------------------------------------------------------
__global__ __launch_bounds__(256) void psm_apply_kernel(
    const float* __restrict__ x, const unsigned char* __restrict__ fm,
    const int* __restrict__ mask, float* __restrict__ out) {
  __shared__ __align__(4) unsigned char s_fm[1024];

  const int tid = threadIdx.x;
  const int bl  = blockIdx.z;
  const int c   = blockIdx.y;
  const int t   = blockIdx.x * 256 + tid;      // float4 index within row
  const bool inb = (t < HWN4);
  const size_t row4 = ((size_t)bl * CDIM + c) * (size_t)HWN4;
  const f32x4* __restrict__ xin  = (const f32x4*)x + row4;
  f32x4* __restrict__ xout       = (f32x4*)out + row4;

  const int l = bl % LDIM;
  if (l == 0) {                          // ego layer: passthrough copy
    if (inb) {
      f32x4 v = __builtin_nontemporal_load(xin + t);
      __builtin_nontemporal_store(v, xout + t);
    }
    return;
  }
  if (mask[bl] == 0) {                   // dropped layer: exact +0.0 fill
    if (inb) {
      f32x4 z = (f32x4)0.0f;
      __builtin_nontemporal_store(z, xout + t);
    }
    return;
  }

  // Stage this block's 1KB mask chunk through LDS with the async copy engine.
  const unsigned byteoff =
      ((unsigned)t * 4u <= (unsigned)(HWN - 4)) ? (unsigned)t * 4u
                                                : (unsigned)(HWN - 4);
  const unsigned char* gsrc = fm + (size_t)bl * HWN + byteoff;
  unsigned mword;
#if __has_builtin(__builtin_amdgcn_global_load_async_to_lds_b32) && \
    __has_builtin(__builtin_amdgcn_s_wait_asynccnt)
  {
    typedef __attribute__((address_space(1))) int as1int;
    typedef __attribute__((address_space(3))) int as3int;
    __builtin_amdgcn_global_load_async_to_lds_b32(
        (as1int*)(void*)gsrc, (as3int*)(void*)(&s_fm[tid * 4]), 0, 0); // ASYNCcnt++
    __builtin_amdgcn_s_wait_asynccnt(0);           // wave reads only own bytes
    mword = *(const unsigned*)(&s_fm[tid * 4]);
  }
#else
  mword = *(const unsigned*)gsrc;                  // fallback: direct load
#endif

  if (inb) {
    f32x4 v = __builtin_nontemporal_load(xin + t);
    v.x *= (float)(mword & 0xffu);
    v.y *= (float)((mword >> 8) & 0xffu);
    v.z *= (float)((mword >> 16) & 0xffu);
    v.w *= (float)((mword >> 24) & 0xffu);
    __builtin_nontemporal_store(v, xout + t);
  }
}

extern "C" void kernel_launch(void* const* d_in, const int* in_sizes, int n_in,
                              void* d_out, int out_size, void* d_ws, size_t ws_size,
                              hipStream_t stream) {
  (void)in_sizes; (void)n_in; (void)out_size; (void)ws_size;
  const float* x   = (const float*)d_in[0];
  const float* psm = (const float*)d_in[1];
  const int*   msk = (const int*)d_in[2];
  float*       out = (float*)d_out;
  unsigned char* fm = (unsigned char*)d_ws;  // needs B*L*HWN = 504000 bytes

  psm_thresh_kernel<<<dim3(BDIM * LDIM), dim3(256), 0, stream>>>(psm, fm);
  psm_apply_kernel<<<dim3(25, CDIM, BDIM * LDIM), dim3(256), 0, stream>>>(
      x, fm, msk, out);
}